// SS3Depth_45243185496331
// MI455X (gfx1250) — compile-verified
//
#include <hip/hip_runtime.h>
#include <math.h>

typedef float v2f __attribute__((ext_vector_type(2)));
typedef float v8f __attribute__((ext_vector_type(8)));

// ---- fixed problem dims (from setup_inputs) ----
#define BB    2
#define DIMS  24
#define CIN   192
#define LL    13824          // 24*24*24
#define DI    384            // d_inner
#define NST   16             // d_state
#define DTR   12             // dt_rank
#define LT    14208          // 384 + 13824
#define CCOF  384            // CC offset in xs
#define NX    768            // 2*d_inner
#define DDEP  1728           // 12*12*12
#define XPC   44             // dt_rank + 2*d_state
#define NCH   64             // scan chunks per channel
#define CLEN  222            // 64 * 222 = 14208

__device__ __forceinline__ float siluf(float x) { return x / (1.0f + __expf(-x)); }

// ---------------------------------------------------------------------------
// Generic fp32 WMMA GEMM:  O(M,N) = X(M,K) @ W(N,K)^T   (M%16==0, N%16==0, K%4==0)
// One wave32 computes a 16x16 tile using V_WMMA_F32_16X16X4_F32.
// ---------------------------------------------------------------------------
__global__ void k_gemm_f32(const float* __restrict__ X, const float* __restrict__ W,
                           float* __restrict__ O, int M, int N, int K) {
    const int lane  = threadIdx.x & 31;
    const int wave  = threadIdx.x >> 5;
    const int tilesN = N >> 4;
    const int tile  = blockIdx.x * (blockDim.x >> 5) + wave;
    const int tm = tile / tilesN, tn = tile % tilesN;
    if (tm * 16 >= M) return;
    const int half = lane >> 4;
    const float* xr = X + (long)(tm * 16 + (lane & 15)) * K + half * 2;
    const float* wr = W + (long)(tn * 16 + (lane & 15)) * K + half * 2;
    v8f acc = {};
    for (int k = 0; k < K; k += 4) {
        v2f a  = *(const v2f*)(xr + k);
        v2f bv = *(const v2f*)(wr + k);
        __builtin_prefetch(xr + k + 128, 0, 1);   // global_prefetch_b8
        acc = __builtin_amdgcn_wmma_f32_16x16x4_f32(false, a, false, bv,
                                                    (short)0, acc, false, false);
    }
    const int nc = tn * 16 + (lane & 15);
    #pragma unroll
    for (int r = 0; r < 8; ++r)
        O[(long)(tm * 16 + r + 8 * half) * N + nc] = acc[r];
}

// ---------------------------------------------------------------------------
// Depthwise conv3d 3x3x3 stride1 pad1 + bias + SiLU.
// Channel index fastest within a wave -> all 27 tap reads of xz are coalesced.
// Output into xs[b, c, 384 + l].
// ---------------------------------------------------------------------------
__global__ void k_conv1(const float* __restrict__ XZ, const float* __restrict__ Wc,
                        const float* __restrict__ bc, float* __restrict__ XS) {
    long idx = (long)blockIdx.x * blockDim.x + threadIdx.x;
    if (idx >= (long)BB * DI * LL) return;
    int c = (int)(idx % DI);
    long r = idx / DI;
    int l = (int)(r % LL);
    int b = (int)(r / LL);
    int w = l % DIMS, h = (l / DIMS) % DIMS, d = l / (DIMS * DIMS);
    const float* wp = Wc + c * 27;
    float s = 0.f;
    for (int kd = 0; kd < 3; ++kd) {
        int dd = d + kd - 1; if (dd < 0 || dd >= DIMS) continue;
        for (int kh = 0; kh < 3; ++kh) {
            int hh = h + kh - 1; if (hh < 0 || hh >= DIMS) continue;
            for (int kw = 0; kw < 3; ++kw) {
                int ww = w + kw - 1; if (ww < 0 || ww >= DIMS) continue;
                long xl = (long)b * LL + (dd * 576 + hh * 24 + ww);
                s += wp[kd * 9 + kh * 3 + kw] * XZ[xl * NX + c];
            }
        }
    }
    s = siluf(s + bc[c]);
    XS[((long)b * DI + c) * LT + CCOF + l] = s;
}

// ---------------------------------------------------------------------------
// Depthwise conv3d 3x3x3 stride2 pad1 + bias.  Reads xs[b,c,384+...], writes dxc.
// ---------------------------------------------------------------------------
__global__ void k_conv2(const float* __restrict__ XS, const float* __restrict__ Wc,
                        const float* __restrict__ bc, float* __restrict__ DXC) {
    long idx = (long)blockIdx.x * blockDim.x + threadIdx.x;
    if (idx >= (long)BB * DI * DDEP) return;
    int lo = (int)(idx % DDEP);
    int t  = (int)(idx / DDEP);
    int c = t % DI, b = t / DI;
    int ow = lo % 12, oh = (lo / 12) % 12, od = lo / 144;
    const float* src = XS + ((long)b * DI + c) * LT + CCOF;
    const float* wp  = Wc + c * 27;
    float s = 0.f;
    for (int kd = 0; kd < 3; ++kd) {
        int id = od * 2 + kd - 1; if (id < 0 || id >= DIMS) continue;
        for (int kh = 0; kh < 3; ++kh) {
            int ih = oh * 2 + kh - 1; if (ih < 0 || ih >= DIMS) continue;
            for (int kw = 0; kw < 3; ++kw) {
                int iw = ow * 2 + kw - 1; if (iw < 0 || iw >= DIMS) continue;
                s += wp[kd * 9 + kh * 3 + kw] * src[id * 576 + ih * 24 + iw];
            }
        }
    }
    DXC[idx] = s + bc[c];
}

// ---------------------------------------------------------------------------
// depth_fc GEMM per batch (WMMA), fused transpose + bias + SiLU into xs[:, :, :384]:
//   fc[b,i,j] = sum_k dxc[b,i,k]*Wfc[j,k] + bias[j];  xs[b, j, i] = silu(fc[b,i,j])
// ---------------------------------------------------------------------------
__global__ void k_depthfc(const float* __restrict__ DXC, const float* __restrict__ Wfc,
                          const float* __restrict__ bias, float* __restrict__ XS) {
    const int b = blockIdx.y;
    const float* A = DXC + (long)b * DI * DDEP;
    const int lane = threadIdx.x & 31;
    const int wave = threadIdx.x >> 5;
    const int tile = blockIdx.x * (blockDim.x >> 5) + wave;
    const int tm = tile / 24, tn = tile % 24;
    if (tm >= 24) return;
    const int half = lane >> 4;
    const float* ar = A   + (long)(tm * 16 + (lane & 15)) * DDEP + half * 2;
    const float* wr = Wfc + (long)(tn * 16 + (lane & 15)) * DDEP + half * 2;
    v8f acc = {};
    for (int k = 0; k < DDEP; k += 4) {
        v2f a  = *(const v2f*)(ar + k);
        v2f bv = *(const v2f*)(wr + k);
        acc = __builtin_amdgcn_wmma_f32_16x16x4_f32(false, a, false, bv,
                                                    (short)0, acc, false, false);
    }
    const int j  = tn * 16 + (lane & 15);
    const float bj = bias[j];
    #pragma unroll
    for (int r = 0; r < 8; ++r) {
        int i = tm * 16 + r + 8 * half;
        XS[((long)b * DI + j) * LT + i] = siluf(acc[r] + bj);   // transposed store
    }
}

// ---------------------------------------------------------------------------
// x_proj (WMMA): x_dbl[b,c,l] = sum_d xs[b,d,l] * Wx[c,d].  M=44 (padded to 48).
// ---------------------------------------------------------------------------
__global__ void k_xproj(const float* __restrict__ XS, const float* __restrict__ Wx,
                        float* __restrict__ XDBL) {
    const int b = blockIdx.y;
    const float* Xb = XS + (long)b * DI * LT;
    const int lane = threadIdx.x & 31;
    const int wave = threadIdx.x >> 5;
    const int tile = blockIdx.x * (blockDim.x >> 5) + wave;
    const int tilesN = LT / 16;                  // 888
    const int tm = tile / tilesN, tn = tile % tilesN;
    if (tm >= 3) return;
    const int half = lane >> 4;
    const int arow = tm * 16 + (lane & 15);
    const int ncol = tn * 16 + (lane & 15);
    const int koff = half * 2;
    v8f acc = {};
    for (int k = 0; k < DI; k += 4) {
        v2f a = {};
        if (arow < XPC) a = *(const v2f*)(Wx + arow * DI + k + koff);
        v2f bv;
        bv.x = Xb[(long)(k + koff) * LT + ncol];
        bv.y = Xb[(long)(k + koff + 1) * LT + ncol];
        acc = __builtin_amdgcn_wmma_f32_16x16x4_f32(false, a, false, bv,
                                                    (short)0, acc, false, false);
    }
    #pragma unroll
    for (int r = 0; r < 8; ++r) {
        int crow = tm * 16 + r + 8 * half;
        if (crow < XPC)
            XDBL[((long)b * XPC + crow) * LT + ncol] = acc[r];
    }
}

// ---------------------------------------------------------------------------
// delta = softplus(dts @ dt_projs_w^T + dt_projs_b)
// ---------------------------------------------------------------------------
__global__ void k_delta(const float* __restrict__ XDBL, const float* __restrict__ dtw,
                        const float* __restrict__ dtb, float* __restrict__ DLT) {
    long idx = (long)blockIdx.x * blockDim.x + threadIdx.x;
    if (idx >= (long)BB * DI * LT) return;
    int l = (int)(idx % LT);
    int t = (int)(idx / LT);
    int d = t % DI, b = t / DI;
    const float* xd = XDBL + (long)b * XPC * LT + l;
    float s = dtb[d];
    #pragma unroll
    for (int r = 0; r < DTR; ++r) s += xd[(long)r * LT] * dtw[d * DTR + r];
    DLT[idx] = (s > 20.f) ? s : log1pf(__expf(s));
}

// ---------------------------------------------------------------------------
// Chunked selective scan.  h_l = exp(dv*A)*h_{l-1} + dv*u*B.  Since
// prod_l exp(dv_l*A_n) = exp(A_n * sum_l dv_l), a chunk's transition is a
// single exp of (A_n * S).  Three passes:
//   P1: per (channel, chunk): zero-state partial c_n and S = sum(dv)  [parallel]
//   P2: per channel: 64-step serial combine -> incoming state per chunk
//   P3: per (channel, chunk): replay with correct h_in, emit outputs  [parallel]
// 16 lanes = 16 states; two independent groups per wave32.
// ---------------------------------------------------------------------------
__global__ void k_scan_p1(const float* __restrict__ DLT, const float* __restrict__ XS,
                          const float* __restrict__ XDBL, const float* __restrict__ Alog,
                          float* __restrict__ CPART, float* __restrict__ SSUM) {
    long tid = (long)blockIdx.x * blockDim.x + threadIdx.x;
    long grp = tid >> 4;                 // = ch*NCH + k
    int  n   = (int)(tid & 15);
    if (grp >= (long)BB * DI * NCH) return;
    int ch = (int)(grp / NCH), k = (int)(grp % NCH);
    int b = ch / DI, d = ch % DI;
    const float Av = -__expf(Alog[d * NST + n]);
    const int l0 = k * CLEN;
    const float* dlt = DLT + (long)ch * LT + l0;
    const float* u   = XS  + (long)ch * LT + l0;
    const float* Bs  = XDBL + ((long)b * XPC + DTR + n) * LT + l0;
    float h = 0.f, S = 0.f;
    for (int l = 0; l < CLEN; ++l) {
        float dv = dlt[l];
        h = __expf(dv * Av) * h + (dv * u[l]) * Bs[l];
        S += dv;
    }
    CPART[grp * 16 + n] = h;
    if (n == 0) SSUM[grp] = S;
}

__global__ void k_scan_p2(const float* __restrict__ CPART, const float* __restrict__ SSUM,
                          const float* __restrict__ Alog, float* __restrict__ HIN) {
    const int lane = threadIdx.x & 31;
    const int half = lane >> 4;
    const int n = lane & 15;
    const int ch = blockIdx.x * 2 + half;        // 0..767
    const int d = ch % DI;
    const float Av = -__expf(Alog[d * NST + n]);
    float h = 0.f;
    for (int k = 0; k < NCH; ++k) {
        long g = (long)ch * NCH + k;
        HIN[g * 16 + n] = h;
        h = __expf(Av * SSUM[g]) * h + CPART[g * 16 + n];
    }
}

__global__ void k_scan_p3(const float* __restrict__ DLT, const float* __restrict__ XS,
                          const float* __restrict__ XDBL, const float* __restrict__ Alog,
                          const float* __restrict__ Ds, const float* __restrict__ HIN,
                          float* __restrict__ YT) {
    long tid = (long)blockIdx.x * blockDim.x + threadIdx.x;
    long grp = tid >> 4;
    int  n   = (int)(tid & 15);
    if (grp >= (long)BB * DI * NCH) return;
    int ch = (int)(grp / NCH), k = (int)(grp % NCH);
    int b = ch / DI, d = ch % DI;
    const float Av = -__expf(Alog[d * NST + n]);
    const float Dd = Ds[d];
    const int l0 = k * CLEN;
    const float* dlt = DLT + (long)ch * LT + l0;
    const float* u   = XS  + (long)ch * LT + l0;
    const float* Bs  = XDBL + ((long)b * XPC + DTR + n) * LT + l0;
    const float* Cs  = XDBL + ((long)b * XPC + DTR + NST + n) * LT + l0;
    float h = HIN[grp * 16 + n];
    for (int l = 0; l < CLEN; ++l) {
        float dv = dlt[l];
        float uv = u[l];
        h = __expf(dv * Av) * h + (dv * uv) * Bs[l];
        float y = h * Cs[l];
        y += __shfl_xor(y, 1, 32);
        y += __shfl_xor(y, 2, 32);
        y += __shfl_xor(y, 4, 32);
        y += __shfl_xor(y, 8, 32);
        int gl = l0 + l;
        if (gl >= CCOF && n == 0)
            YT[((long)b * LL + (gl - CCOF)) * DI + d] = y + Dd * uv;
    }
}

// ---------------------------------------------------------------------------
// LayerNorm over d_inner (in place on YT), then * silu(z).  One block per row.
// ---------------------------------------------------------------------------
__global__ void k_ln(float* __restrict__ YT, const float* __restrict__ XZ,
                     const float* __restrict__ g, const float* __restrict__ be) {
    __shared__ float red[128];
    const int row = blockIdx.x;
    float* y = YT + (long)row * DI;
    const float* z = XZ + (long)row * NX + DI;
    float v[3];
    float s = 0.f, s2 = 0.f;
    #pragma unroll
    for (int i = 0; i < 3; ++i) {
        float t = y[threadIdx.x + 128 * i];
        v[i] = t; s += t; s2 += t * t;
    }
    red[threadIdx.x] = s; __syncthreads();
    for (int off = 64; off > 0; off >>= 1) {
        if (threadIdx.x < off) red[threadIdx.x] += red[threadIdx.x + off];
        __syncthreads();
    }
    const float mean = red[0] * (1.f / DI);
    __syncthreads();
    red[threadIdx.x] = s2; __syncthreads();
    for (int off = 64; off > 0; off >>= 1) {
        if (threadIdx.x < off) red[threadIdx.x] += red[threadIdx.x + off];
        __syncthreads();
    }
    const float var  = red[0] * (1.f / DI) - mean * mean;
    const float rstd = rsqrtf(var + 1e-5f);
    #pragma unroll
    for (int i = 0; i < 3; ++i) {
        int dd = threadIdx.x + 128 * i;
        float t = (v[i] - mean) * rstd * g[dd] + be[dd];
        y[dd] = t * siluf(z[dd]);
    }
}

// ---------------------------------------------------------------------------
extern "C" void kernel_launch(void* const* d_in, const int* in_sizes, int n_in,
                              void* d_out, int out_size, void* d_ws, size_t ws_size,
                              hipStream_t stream) {
    const float* x         = (const float*)d_in[0];
    const float* in_proj_w = (const float*)d_in[1];
    const float* conv3d_w  = (const float*)d_in[2];
    const float* conv3d_b  = (const float*)d_in[3];
    const float* dconv_w   = (const float*)d_in[4];
    const float* dconv_b   = (const float*)d_in[5];
    const float* dfc_w     = (const float*)d_in[6];
    const float* dfc_b     = (const float*)d_in[7];
    const float* xproj_w   = (const float*)d_in[8];
    const float* dt_w      = (const float*)d_in[9];
    const float* dt_b      = (const float*)d_in[10];
    const float* A_logs    = (const float*)d_in[11];
    const float* Ds        = (const float*)d_in[12];
    const float* ln_g      = (const float*)d_in[13];
    const float* ln_b      = (const float*)d_in[14];
    const float* out_w     = (const float*)d_in[15];
    float* out = (float*)d_out;

    // workspace layout (floats)
    float* ws    = (float*)d_ws;
    float* xz    = ws;                                  // (B*L, 768)        21,233,664
    float* xs    = xz    + (long)BB * LL * NX;          // (B, 384, 14208)   10,911,744
    float* dxc   = xs    + (long)BB * DI * LT;          // (B, 384, 1728)     1,327,104
    float* xdbl  = dxc   + (long)BB * DI * DDEP;        // (B, 44, 14208)     1,250,304
    float* dlt   = xdbl  + (long)BB * XPC * LT;         // (B, 384, 14208)   10,911,744
    float* yt    = dlt   + (long)BB * DI * LT;          // (B*L, 384)        10,616,832
    float* cpart = yt    + (long)BB * LL * DI;          // (768*64, 16)         786,432
    float* ssum  = cpart + (long)BB * DI * NCH * NST;   // (768*64)              49,152
    float* hin   = ssum  + (long)BB * DI * NCH;         // (768*64, 16)         786,432

    const int M1 = BB * LL;                             // 27648

    // 1) in_proj GEMM: xz = x @ in_proj_w^T   (27648 x 768, K=192)
    {
        int tiles = (M1 / 16) * (NX / 16);              // 82944
        k_gemm_f32<<<tiles / 8, 256, 0, stream>>>(x, in_proj_w, xz, M1, NX, CIN);
    }
    // 2) depthwise conv 3x3x3 s1 + SiLU -> xs[:, :, 384:]
    {
        long n = (long)BB * DI * LL;
        k_conv1<<<(int)((n + 255) / 256), 256, 0, stream>>>(xz, conv3d_w, conv3d_b, xs);
    }
    // 3) depthwise conv 3x3x3 s2 -> dxc
    {
        long n = (long)BB * DI * DDEP;
        k_conv2<<<(int)((n + 255) / 256), 256, 0, stream>>>(xs, dconv_w, dconv_b, dxc);
    }
    // 4) depth_fc GEMM + transpose + SiLU -> xs[:, :, :384]
    {
        dim3 grid(72, BB);                              // 576 tiles / 8 waves
        k_depthfc<<<grid, 256, 0, stream>>>(dxc, dfc_w, dfc_b, xs);
    }
    // 5) x_proj GEMM -> xdbl (B, 44, 14208)
    {
        dim3 grid(333, BB);                             // 3*888 tiles / 8 waves
        k_xproj<<<grid, 256, 0, stream>>>(xs, xproj_w, xdbl);
    }
    // 6) delta = softplus(dts @ dt_w^T + dt_b)
    {
        long n = (long)BB * DI * LT;
        k_delta<<<(int)((n + 255) / 256), 256, 0, stream>>>(xdbl, dt_w, dt_b, dlt);
    }
    // 7) chunked selective scan -> yt (B*L, 384)
    {
        long p13_threads = (long)BB * DI * NCH * NST;   // 786432
        int  p13_blocks  = (int)((p13_threads + 255) / 256);
        k_scan_p1<<<p13_blocks, 256, 0, stream>>>(dlt, xs, xdbl, A_logs, cpart, ssum);
        k_scan_p2<<<BB * DI / 2, 32, 0, stream>>>(cpart, ssum, A_logs, hin);
        k_scan_p3<<<p13_blocks, 256, 0, stream>>>(dlt, xs, xdbl, A_logs, Ds, hin, yt);
    }
    // 8) layernorm * silu(z), in place on yt
    k_ln<<<M1, 128, 0, stream>>>(yt, xz, ln_g, ln_b);
    // 9) out_proj GEMM: out = yt @ out_proj_w^T   (27648 x 192, K=384)
    {
        int tiles = (M1 / 16) * (CIN / 16);             // 20736
        k_gemm_f32<<<tiles / 8, 256, 0, stream>>>(yt, out_w, out, M1, CIN, DI);
    }
}